// GatherModel_11879879543385
// MI455X (gfx1250) — compile-verified
//
#include <hip/hip_runtime.h>

// ---------------------------------------------------------------------------
// y_{t+1}[b,n,c] = lambda * sum_{k<16} w[k,n] * y_t[b, idx[k,n], c],  c<256
// 5 iterations, ping-pong between d_out and d_ws (odd count ends in d_out).
//
// Reduction over k is done with V_WMMA_F32_16X16X4_F32:
//   A (16x4 f32, documented layout): lanes 0-15: V0=A[m,0],V1=A[m,1];
//                                    lanes16-31: V0=A[m,2],V1=A[m,3]
//     -> A[m,kk] = gathered y[b, idx[k0+kk], e0+m]   (coalesced 64B per row)
//   B (4x16 f32): row kk broadcast across all 16 columns = lambda*w[k0+kk]
//     (assumed row->lane-half mapping mirrors A: rows {0,2} in V0, {1,3} in V1)
//   D[m,*] accumulates the weighted sum for element e0+m, replicated in cols.
// Memory traffic identical to the optimal scalar gather-FMA version.
// ---------------------------------------------------------------------------

typedef float v2f __attribute__((ext_vector_type(2)));
typedef float v8f __attribute__((ext_vector_type(8)));

#define BATCH 8
#define NDIM  2048
#define KDIM  16
#define CDIM  256   // D1*D2 = 4*64

__global__ __launch_bounds__(128)
void gather_wsum_wmma_kernel(const float* __restrict__ src,   // (B, N, 256)
                             float* __restrict__ dst,         // (B, N, 256)
                             const int* __restrict__ ind,     // (K, N)
                             const float* __restrict__ wgt,   // (K, N)
                             const float* __restrict__ lambda_p)
{
    const int n    = blockIdx.x;          // 0..N-1
    const int b    = blockIdx.y;          // 0..B-1
    const int tid  = threadIdx.x;
    const int lane = tid & 31;
    const int wave = tid >> 5;            // 4 waves per block
    const int half = lane >> 4;           // 0: lanes 0-15, 1: lanes 16-31
    const int m    = lane & 15;           // row index within A / element offset

    const float lambda = lambda_p[0];

    // Per-n gather offsets (in floats) and lambda-folded weights. Uniform
    // addresses -> scalar loads; kept in registers, indexed only by constants.
    int   off[KDIM];
    float wk[KDIM];
#pragma unroll
    for (int k = 0; k < KDIM; ++k) {
        off[k] = ind[k * NDIM + n] * CDIM;
        wk[k]  = lambda * wgt[k * NDIM + n];
    }

    const float* __restrict__ srcb = src + (size_t)b * (NDIM * CDIM);
    float*       __restrict__ outp = dst + ((size_t)b * NDIM + n) * CDIM;

    // Each wave produces 4 chunks of 16 output elements -> 4*4*16 = 256 = CDIM.
#pragma unroll
    for (int c = 0; c < 4; ++c) {
        const int e0 = (wave * 4 + c) * 16;
        v8f acc = {};
#pragma unroll
        for (int k0 = 0; k0 < KDIM; k0 += 4) {
            // Lane-half selects K rows {k0,k0+1} (lanes 0-15) or {k0+2,k0+3}.
            const int   oa = half ? off[k0 + 2] : off[k0 + 0];
            const int   ob = half ? off[k0 + 3] : off[k0 + 1];
            const float wa = half ? wk[k0 + 2]  : wk[k0 + 0];
            const float wb = half ? wk[k0 + 3]  : wk[k0 + 1];

            v2f a, bm;
            a.x  = srcb[(size_t)oa + e0 + m];   // contiguous across lanes 0-15
            a.y  = srcb[(size_t)ob + e0 + m];
            bm.x = wa;                          // broadcast over 16 columns
            bm.y = wb;

            acc = __builtin_amdgcn_wmma_f32_16x16x4_f32(
                false, a, false, bm, (short)0, acc, false, false);
        }
        // D layout: VGPR r holds M=r (lanes 0-15) and M=r+8 (lanes 16-31);
        // all 16 columns identical -> lane 0 / lane 16 hold the full result.
        if (m == 0) {
            float* p = outp + e0 + half * 8;
#pragma unroll
            for (int r = 0; r < 8; ++r) p[r] = acc[r];
        }
    }
}

extern "C" void kernel_launch(void* const* d_in, const int* in_sizes, int n_in,
                              void* d_out, int out_size, void* d_ws, size_t ws_size,
                              hipStream_t stream) {
    const float* inputs = (const float*)d_in[0];   // (B,N,4,64) f32
    const int*   ind1   = (const int*)d_in[1];     // (K,N,1)
    const float* w1     = (const float*)d_in[2];   // (K,N,1,1,1)
    const float* lam    = (const float*)d_in[3];   // scalar (1 elem)

    float* out = (float*)d_out;
    float* tmp = (float*)d_ws;                     // needs 16 MB scratch

    dim3 grid(NDIM, BATCH);
    dim3 block(128);

    // 5 iterations: in->out, out->ws, ws->out, out->ws, ws->out (ends in d_out)
    const float* srcs[5] = { inputs, out, tmp, out, tmp };
    float*       dsts[5] = { out,    tmp, out, tmp, out };
    for (int it = 0; it < 5; ++it) {
        gather_wsum_wmma_kernel<<<grid, block, 0, stream>>>(
            srcs[it], dsts[it], ind1, w1, lam);
    }
}